// GCN_67568425501456
// MI455X (gfx1250) — compile-verified
//
#include <hip/hip_runtime.h>

// 3-layer GCN for MI455X (gfx1250, wave32).
//   per layer: T = X @ W  (WMMA f32 16x16x4, exact fp32)
//              OUT = T*dinv^2 + b          (self loop + bias init)
//              OUT[dst] += T[src]*dinv[src]*dinv[dst]  (edge scatter, HW f32 atomics)
//   ReLU fused into next layer's GEMM A-tile load.

typedef __attribute__((ext_vector_type(2))) float v2f;
typedef __attribute__((ext_vector_type(8))) float v8f;

#define D 128   // D_IN = D_HID = D_OUT = 128

// ---------------- normalization ----------------

__global__ void zero_f32_kernel(float* __restrict__ p, int n) {
  int i = blockIdx.x * blockDim.x + threadIdx.x;
  if (i < n) p[i] = 0.0f;
}

__global__ void count_deg_kernel(const long long* __restrict__ dst,
                                 float* __restrict__ deg, int E) {
  int e = blockIdx.x * blockDim.x + threadIdx.x;
  if (e < E) unsafeAtomicAdd(&deg[dst[e]], 1.0f);   // global_atomic_add_f32
}

__global__ void dinv_kernel(float* __restrict__ deg, int n) {
  int i = blockIdx.x * blockDim.x + threadIdx.x;
  if (i < n) {
    float d = deg[i] + 1.0f;          // + self loop, so d >= 1 always
    deg[i] = 1.0f / sqrtf(d);         // in-place: deg buffer becomes dinv
  }
}

// ---------------- dense transform: T = (relu?)X @ W ----------------
// Block: 256 threads = 8 wave32. Block handles 16 rows x 128 cols.
// A tile (16x128) staged in LDS, shared by all 8 waves; each wave computes a
// 16x16 output tile with 32 chained v_wmma_f32_16x16x4_f32 over K=128.

__global__ __launch_bounds__(256)
void gemm_wmma_kernel(const float* __restrict__ X, const float* __restrict__ W,
                      float* __restrict__ T, int apply_relu) {
  __shared__ float As[16 * D];

  const int    tid  = threadIdx.x;
  const size_t row0 = (size_t)blockIdx.x * 16;   // N = 100000 is a multiple of 16

  // Cooperative A-tile load (2048 floats as 512 float4s), optional fused ReLU.
  const float4* xsrc = (const float4*)(X + row0 * D);
  float4*       adst = (float4*)As;
  for (int i = tid; i < 16 * D / 4; i += 256) {
    float4 v = xsrc[i];
    if (apply_relu) {
      v.x = fmaxf(v.x, 0.0f); v.y = fmaxf(v.y, 0.0f);
      v.z = fmaxf(v.z, 0.0f); v.w = fmaxf(v.w, 0.0f);
    }
    adst[i] = v;
  }
  __syncthreads();

  const int wave = tid >> 5;          // 0..7 -> 16-col tile
  const int lane = tid & 31;
  const int col0 = wave * 16;
  const int m    = lane & 15;         // A: row / B: col held by this lane
  const int koff = (lane >> 4) << 1;  // lanes 0-15: K+0,K+1 ; lanes 16-31: K+2,K+3

  v8f acc = {};
  #pragma unroll
  for (int k = 0; k < D; k += 4) {
    v2f a, b;
    // A-matrix 16x4 fp32 fragment (ISA layout: VGPR0=K0/K2, VGPR1=K1/K3)
    a.x = As[m * D + k + koff];
    a.y = As[m * D + k + koff + 1];
    // B-matrix 4x16 fp32 fragment (rows striped across lanes)
    b.x = W[(k + koff)     * D + col0 + m];
    b.y = W[(k + koff + 1) * D + col0 + m];
    acc = __builtin_amdgcn_wmma_f32_16x16x4_f32(
        /*neg_a=*/false, a, /*neg_b=*/false, b,
        /*c_mod=*/(short)0, acc, /*reuse_a=*/false, /*reuse_b=*/false);
  }

  // C/D layout: VGPR r -> (M = r + 8*(lane>=16), N = lane&15) within tile.
  const int ncol  = col0 + m;
  const int mbase = (lane >> 4) * 8;
  float* trow = T + (row0 + mbase) * D + ncol;
  #pragma unroll
  for (int r = 0; r < 8; ++r)
    trow[(size_t)r * D] = acc[r];
}

// ---------------- self-loop + bias init: OUT = T*dinv^2 + b ----------------

__global__ __launch_bounds__(256)
void self_bias_kernel(const float* __restrict__ T, const float* __restrict__ dinv,
                      const float* __restrict__ bias, float* __restrict__ out, int n) {
  int idx = blockIdx.x * blockDim.x + threadIdx.x;     // one float4 per thread
  if (idx >= n * (D / 4)) return;
  int row = idx >> 5;                                  // D/4 == 32
  int c4  = idx & 31;
  float w = dinv[row];
  w *= w;                                              // self-loop norm = dinv^2
  float4 t  = ((const float4*)T)[idx];
  float4 bb = ((const float4*)bias)[c4];
  float4 o;
  o.x = t.x * w + bb.x;  o.y = t.y * w + bb.y;
  o.z = t.z * w + bb.z;  o.w = t.w * w + bb.w;
  ((float4*)out)[idx] = o;
}

// ---------------- edge scatter: OUT[dst] += T[src] * dinv[src]*dinv[dst] ----
// One wave32 per edge; lane i owns float4 i of the 128-wide row.

__global__ __launch_bounds__(256)
void scatter_kernel(const long long* __restrict__ src, const long long* __restrict__ dst,
                    const float* __restrict__ dinv, const float* __restrict__ T,
                    float* __restrict__ out, int E) {
  int gw   = (blockIdx.x * 256 + threadIdx.x) >> 5;
  int lane = threadIdx.x & 31;
  if (gw >= E) return;
  long long s = src[gw];
  long long d = dst[gw];
  float w = dinv[s] * dinv[d];
  float4 m4 = ((const float4*)(T + (size_t)s * D))[lane];   // coalesced 512B row
  float* orow = out + (size_t)d * D + lane * 4;
  unsafeAtomicAdd(orow + 0, m4.x * w);
  unsafeAtomicAdd(orow + 1, m4.y * w);
  unsafeAtomicAdd(orow + 2, m4.z * w);
  unsafeAtomicAdd(orow + 3, m4.w * w);
}

// ---------------- driver ----------------

extern "C" void kernel_launch(void* const* d_in, const int* in_sizes, int n_in,
                              void* d_out, int out_size, void* d_ws, size_t ws_size,
                              hipStream_t stream) {
  (void)n_in; (void)out_size; (void)ws_size;

  const float*     x    = (const float*)d_in[0];
  const long long* edge = (const long long*)d_in[1];   // int64 [2, E] flattened
  const float*     W0   = (const float*)d_in[2];
  const float*     b0   = (const float*)d_in[3];
  const float*     W1   = (const float*)d_in[4];
  const float*     b1   = (const float*)d_in[5];
  const float*     W2   = (const float*)d_in[6];
  const float*     b2   = (const float*)d_in[7];
  float*           out  = (float*)d_out;

  const int N = in_sizes[0] / D;       // 100000
  const int E = in_sizes[1] / 2;       // 1600000
  const long long* src = edge;         // edge_index[0]
  const long long* dst = edge + E;     // edge_index[1]

  // Workspace layout: dinv[N] | T[N*128] | AGG[N*128]  (~103 MB)
  float* dinv = (float*)d_ws;
  size_t off  = (((size_t)N * sizeof(float)) + 255) & ~(size_t)255;
  float* T    = (float*)((char*)d_ws + off);
  float* AGG  = T + (size_t)N * D;

  const dim3 b256(256);
  const int  nb_n    = (N + 255) / 256;
  const int  nb_e    = (E + 255) / 256;
  const int  gemm_nb = N / 16;                  // exact: 100000 % 16 == 0
  const int  sb_nb   = (N * (D / 4) + 255) / 256;
  const int  sc_nb   = (E + 7) / 8;             // 8 waves (edges) per block

  // Symmetric normalization: dinv = 1/sqrt(deg + 1)
  zero_f32_kernel <<<nb_n, b256, 0, stream>>>(dinv, N);
  count_deg_kernel<<<nb_e, b256, 0, stream>>>(dst, dinv, E);
  dinv_kernel     <<<nb_n, b256, 0, stream>>>(dinv, N);

  // Layer 0
  gemm_wmma_kernel<<<gemm_nb, b256, 0, stream>>>(x,   W0, T, 0);
  self_bias_kernel<<<sb_nb,   b256, 0, stream>>>(T, dinv, b0, AGG, N);
  scatter_kernel  <<<sc_nb,   b256, 0, stream>>>(src, dst, dinv, T, AGG, E);

  // Layer 1 (ReLU of AGG fused into GEMM A-load)
  gemm_wmma_kernel<<<gemm_nb, b256, 0, stream>>>(AGG, W1, T, 1);
  self_bias_kernel<<<sb_nb,   b256, 0, stream>>>(T, dinv, b1, AGG, N);
  scatter_kernel  <<<sc_nb,   b256, 0, stream>>>(src, dst, dinv, T, AGG, E);

  // Layer 2 -> d_out (no final ReLU)
  gemm_wmma_kernel<<<gemm_nb, b256, 0, stream>>>(AGG, W2, T, 1);
  self_bias_kernel<<<sb_nb,   b256, 0, stream>>>(T, dinv, b2, out, N);
  scatter_kernel  <<<sc_nb,   b256, 0, stream>>>(src, dst, dinv, T, out, E);
}